// UnifiedAlignmentModule_13108240188112
// MI455X (gfx1250) — compile-verified
//
#include <hip/hip_runtime.h>

// CDNA5 / gfx1250 fused kernel:
//   out = sigmoid(0.5) * (Xs @ Ws^T + bs) + (1 - sigmoid(0.5)) * (Xl @ Wl^T + bl)
// Uses V_WMMA_F32_16X16X4_F32 (native fp32 matrix op, wave32).

typedef float v2f __attribute__((ext_vector_type(2)));
typedef float v8f __attribute__((ext_vector_type(8)));

#define NROWS  4096
#define DDIM   512
#define MTILES 4                       // 16x16 tiles stacked in M per wave -> 64x16 tile
#define WAVES_PER_BLOCK 4

__global__ __launch_bounds__(WAVES_PER_BLOCK * 32)
void fused_dual_gemm_combine(const float* __restrict__ Xs,
                             const float* __restrict__ Xl,
                             const float* __restrict__ Ws,
                             const float* __restrict__ bs,
                             const float* __restrict__ Wl,
                             const float* __restrict__ bl,
                             float* __restrict__ out)
{
    const int wave   = blockIdx.x * WAVES_PER_BLOCK + (threadIdx.x >> 5);
    const int nTiles = DDIM / 16;                 // 32 tiles along N
    const int mGroup = wave / nTiles;             // 0..63
    const int jTile  = wave % nTiles;             // 0..31
    const int ibase  = mGroup * (16 * MTILES);
    const int jbase  = jTile * 16;

    const int lane = threadIdx.x & 31;
    const int half = lane >> 4;                   // selects K-pair within fragment
    const int l15  = lane & 15;

    v8f accS[MTILES];
    v8f accL[MTILES];
#pragma unroll
    for (int t = 0; t < MTILES; ++t) { accS[t] = {}; accL[t] = {}; }

    // B fragment source: B[k][n] = W[jbase+n][k]  (W row-major, contiguous in k)
    const int    colW  = jbase + l15;
    const float* wsRow = Ws + (size_t)colW * DDIM;
    const float* wlRow = Wl + (size_t)colW * DDIM;

    for (int k0 = 0; k0 < DDIM; k0 += 4) {
        const int kb = k0 + half * 2;             // 8B aligned (kb even)
        const v2f bS = *(const v2f*)(wsRow + kb);
        const v2f bL = *(const v2f*)(wlRow + kb);
#pragma unroll
        for (int t = 0; t < MTILES; ++t) {
            const int row = ibase + t * 16 + l15;
            const v2f aS = *(const v2f*)(Xs + (size_t)row * DDIM + kb);
            const v2f aL = *(const v2f*)(Xl + (size_t)row * DDIM + kb);
            // (neg_a, A, neg_b, B, c_mod, C, reuse_a, reuse_b)
            accS[t] = __builtin_amdgcn_wmma_f32_16x16x4_f32(
                false, aS, false, bS, (short)0, accS[t], false, false);
            accL[t] = __builtin_amdgcn_wmma_f32_16x16x4_f32(
                false, aL, false, bL, (short)0, accL[t], false, false);
        }
    }

    const float aw = 0.62245933120185459f;        // sigmoid(0.5)
    const float biasS = bs[colW];
    const float biasL = bl[colW];
#pragma unroll
    for (int t = 0; t < MTILES; ++t) {
#pragma unroll
        for (int v = 0; v < 8; ++v) {
            const int row = ibase + t * 16 + half * 8 + v;   // C/D VGPR layout
            out[(size_t)row * DDIM + colW] =
                aw * (accS[t][v] + biasS) + (1.0f - aw) * (accL[t][v] + biasL);
        }
    }
}

extern "C" void kernel_launch(void* const* d_in, const int* in_sizes, int n_in,
                              void* d_out, int out_size, void* d_ws, size_t ws_size,
                              hipStream_t stream) {
    const float* Xs = (const float*)d_in[0];   // small_graph_features [4096,512]
    const float* Xl = (const float*)d_in[1];   // large_graph_features [4096,512]
    const float* Ws = (const float*)d_in[2];   // W_small [512,512]
    const float* bs = (const float*)d_in[3];   // b_small [512]
    const float* Wl = (const float*)d_in[4];   // W_large [512,512]
    const float* bl = (const float*)d_in[5];   // b_large [512]
    float* out = (float*)d_out;                // [4096,512] fp32

    const int totalWaves = (NROWS / (16 * MTILES)) * (DDIM / 16);  // 2048
    const int blocks = totalWaves / WAVES_PER_BLOCK;               // 512
    fused_dual_gemm_combine<<<blocks, WAVES_PER_BLOCK * 32, 0, stream>>>(
        Xs, Xl, Ws, bs, Wl, bl, out);
}